// PAM_Module_32229434589618
// MI455X (gfx1250) — compile-verified
//
#include <hip/hip_runtime.h>

typedef float v2f __attribute__((ext_vector_type(2)));
typedef float v8f __attribute__((ext_vector_type(8)));

#define Bsz 8
#define Cch 512
#define Nsp 4096
#define Dd  64
#define PROW 160   // padded pair-row stride (floats): 128 data + 32 pad
                   // -> p and p+1 rows land in disjoint LDS bank halves

__device__ __forceinline__ float softplus_f(float x) {
    return (x > 20.0f) ? x : log1pf(__expf(x));
}

// ---------------------------------------------------------------------------
// Kernel 1: fused QKV projection.
// GEMM per batch: W' [640,512] x xf [512,4096] -> [640,4096].
// Rows [0,64)=Q (softplus, [B,N,d]), [64,128)=K (softplus, [B,d,N]),
// [128,640)=V ([B,C,N]).
// Block = 4 waves, each owning a 16(o) x 64(n) strip (4 accumulators).
// xf tile (64k x 64n) is double-buffered in LDS using a pair-interleaved
// layout: element (k,n) lives at Bs[(k>>1)*PROW + 2n + (k&1)], so a WMMA
// B-fragment {B[k][n], B[k+1][n]} is ONE aligned ds_load_b64, no repack.
// ---------------------------------------------------------------------------
__global__ void __launch_bounds__(128) qkv_kernel(
    const float* __restrict__ X,
    const float* __restrict__ Wq, const float* __restrict__ bq,
    const float* __restrict__ Wk, const float* __restrict__ bk,
    const float* __restrict__ Wv, const float* __restrict__ bv,
    float* __restrict__ Qb, float* __restrict__ Kb, float* __restrict__ Vb)
{
    __shared__ float Bs[2][32 * PROW];          // 2 x 20 KB

    const int tid  = threadIdx.x;
    const int wave = tid >> 5;
    const int lane = tid & 31;
    const int half = lane >> 4;
    const int l15  = lane & 15;

    int bid = blockIdx.x;
    const int tO4 = bid % 10; bid /= 10;        // 40 o-tiles / 4 waves
    const int tN  = bid % 64;                   // 4096 / 64
    const int b   = bid / 64;

    const int o0 = (tO4 * 4 + wave) * 16;
    const int n0 = tN * 64;

    const float* Wsel; const float* bsel; int osel; int mode;
    if (o0 < 64)       { Wsel = Wq; bsel = bq; osel = o0;       mode = 0; }
    else if (o0 < 128) { Wsel = Wk; bsel = bk; osel = o0 - 64;  mode = 1; }
    else               { Wsel = Wv; bsel = bv; osel = o0 - 128; mode = 2; }

    const float* Ap = Wsel + (size_t)(osel + l15) * Cch;
    const float* Xb = X + (size_t)b * Cch * Nsp;

    v8f acc[4];
    #pragma unroll
    for (int s = 0; s < 4; ++s) acc[s] = (v8f){0.f,0.f,0.f,0.f,0.f,0.f,0.f,0.f};

    // staging regs: 4 blocks of (2 rows x 4 cols) per thread per chunk
    float4 st0[4], st1[4];

    // prologue: stage + commit chunk 0
    #pragma unroll
    for (int j = 0; j < 4; ++j) {
        int blk = tid + j * 128;                // 0..511
        int p   = blk >> 4;                     // pair-row 0..31
        int cq  = (blk & 15) * 4;               // col quad
        st0[j] = *reinterpret_cast<const float4*>(Xb + (size_t)(2*p    ) * Nsp + n0 + cq);
        st1[j] = *reinterpret_cast<const float4*>(Xb + (size_t)(2*p + 1) * Nsp + n0 + cq);
    }
    #pragma unroll
    for (int j = 0; j < 4; ++j) {
        int blk = tid + j * 128;
        int p   = blk >> 4;
        int cq  = (blk & 15) * 4;
        float* dst = &Bs[0][p * PROW + cq * 2];
        *reinterpret_cast<float4*>(dst)     = make_float4(st0[j].x, st1[j].x, st0[j].y, st1[j].y);
        *reinterpret_cast<float4*>(dst + 4) = make_float4(st0[j].z, st1[j].z, st0[j].w, st1[j].w);
    }
    __syncthreads();

    const int coff = l15 * 2;                   // lane column offset (pair units)

    for (int i = 0; i < 8; ++i) {               // 8 chunks of K=64
        const int kc = i * 64;

        if (i < 7) {                            // issue next chunk's loads now
            #pragma unroll
            for (int j = 0; j < 4; ++j) {
                int blk = tid + j * 128;
                int p   = blk >> 4;
                int cq  = (blk & 15) * 4;
                st0[j] = *reinterpret_cast<const float4*>(
                    Xb + (size_t)(kc + 64 + 2*p    ) * Nsp + n0 + cq);
                st1[j] = *reinterpret_cast<const float4*>(
                    Xb + (size_t)(kc + 64 + 2*p + 1) * Nsp + n0 + cq);
            }
        }

        // A fragments for whole chunk (16 pipelined global b64 loads)
        v2f areg[16];
        #pragma unroll
        for (int kk = 0; kk < 16; ++kk)
            areg[kk] = *reinterpret_cast<const v2f*>(Ap + kc + kk * 4 + 2 * half);

        const float* Bcur = Bs[i & 1];
        v2f bfrag[4];
        #pragma unroll
        for (int s = 0; s < 4; ++s)             // kk = 0 -> pair-row = half
            bfrag[s] = *reinterpret_cast<const v2f*>(&Bcur[half * PROW + s * 32 + coff]);

        #pragma unroll
        for (int kk = 0; kk < 16; ++kk) {
            v2f bnext[4];
            if (kk < 15) {
                const int pn = (kk + 1) * 2 + half;
                #pragma unroll
                for (int s = 0; s < 4; ++s)
                    bnext[s] = *reinterpret_cast<const v2f*>(&Bcur[pn * PROW + s * 32 + coff]);
            }
            #pragma unroll
            for (int s = 0; s < 4; ++s)
                acc[s] = __builtin_amdgcn_wmma_f32_16x16x4_f32(
                    false, areg[kk], false, bfrag[s], (short)0, acc[s], false, false);
            if (kk < 15) {
                #pragma unroll
                for (int s = 0; s < 4; ++s) bfrag[s] = bnext[s];
            }
        }

        __syncthreads();
        if (i < 7) {
            #pragma unroll
            for (int j = 0; j < 4; ++j) {
                int blk = tid + j * 128;
                int p   = blk >> 4;
                int cq  = (blk & 15) * 4;
                float* dst = &Bs[(i + 1) & 1][p * PROW + cq * 2];
                *reinterpret_cast<float4*>(dst)     = make_float4(st0[j].x, st1[j].x, st0[j].y, st1[j].y);
                *reinterpret_cast<float4*>(dst + 4) = make_float4(st0[j].z, st1[j].z, st0[j].w, st1[j].w);
            }
        }
        __syncthreads();
    }

    float bias8[8];
    #pragma unroll
    for (int v = 0; v < 8; ++v) bias8[v] = bsel[osel + 8 * half + v];

    if (mode == 0) {
        #pragma unroll
        for (int s = 0; s < 4; ++s) {
            int n = n0 + s * 16 + l15;
            float* qp = Qb + ((size_t)b * Nsp + n) * Dd + osel + 8 * half;
            float r[8];
            #pragma unroll
            for (int v = 0; v < 8; ++v) r[v] = softplus_f(acc[s][v] + bias8[v]);
            *reinterpret_cast<float4*>(qp)     = make_float4(r[0], r[1], r[2], r[3]);
            *reinterpret_cast<float4*>(qp + 4) = make_float4(r[4], r[5], r[6], r[7]);
        }
    } else if (mode == 1) {
        #pragma unroll
        for (int v = 0; v < 8; ++v) {
            int sr = osel + 8 * half + v;
            #pragma unroll
            for (int s = 0; s < 4; ++s) {
                int n = n0 + s * 16 + l15;
                Kb[((size_t)b * Dd + sr) * Nsp + n] = softplus_f(acc[s][v] + bias8[v]);
            }
        }
    } else {
        #pragma unroll
        for (int v = 0; v < 8; ++v) {
            int sr = osel + 8 * half + v;
            #pragma unroll
            for (int s = 0; s < 4; ++s) {
                int n = n0 + s * 16 + l15;
                Vb[((size_t)b * Cch + sr) * Nsp + n] = acc[s][v] + bias8[v];
            }
        }
    }
}

// ---------------------------------------------------------------------------
// Kernel 2: KV aggregate. KV[m][c] = sum_n K[m][n] * V[c][n] per batch.
// Wave = 16(m) x 64(c). All fragment loads are contiguous global b64
// (contraction dim n is the fast axis of both K and V). The 16-k batch of
// loads is register double-buffered so each batch's loads overlap the
// previous batch's 16 WMMAs.
// ---------------------------------------------------------------------------
__global__ void __launch_bounds__(128) kv_kernel(
    const float* __restrict__ Kb, const float* __restrict__ Vb,
    float* __restrict__ KVb)
{
    const int tid  = threadIdx.x;
    const int wave = tid >> 5;
    const int lane = tid & 31;
    const int half = lane >> 4;
    const int l15  = lane & 15;

    int wid = blockIdx.x * 4 + wave;
    const int cg = wid % 8; wid /= 8;           // 512 / 64
    const int tM = wid % 4;                     // 64 / 16
    const int b  = wid / 4;
    const int m0 = tM * 16;
    const int c0 = cg * 64;

    const float* Arow = Kb + ((size_t)b * Dd + m0 + l15) * Nsp + 2 * half;
    const float* Brow[4];
    #pragma unroll
    for (int s = 0; s < 4; ++s)
        Brow[s] = Vb + ((size_t)b * Cch + c0 + s * 16 + l15) * Nsp + 2 * half;

    v8f acc[4];
    #pragma unroll
    for (int s = 0; s < 4; ++s) acc[s] = (v8f){0.f,0.f,0.f,0.f,0.f,0.f,0.f,0.f};

    v2f a[2][4], bv[2][4][4];
    #pragma unroll
    for (int i = 0; i < 4; ++i)
        a[0][i] = *reinterpret_cast<const v2f*>(Arow + i * 4);
    #pragma unroll
    for (int s = 0; s < 4; ++s)
        #pragma unroll
        for (int i = 0; i < 4; ++i)
            bv[0][s][i] = *reinterpret_cast<const v2f*>(Brow[s] + i * 4);

    #pragma unroll 1
    for (int k0 = 0; k0 < Nsp; k0 += 16) {
        const int cur = (k0 >> 4) & 1, nxt = cur ^ 1;
        if (k0 + 16 < Nsp) {
            #pragma unroll
            for (int i = 0; i < 4; ++i)
                a[nxt][i] = *reinterpret_cast<const v2f*>(Arow + k0 + 16 + i * 4);
            #pragma unroll
            for (int s = 0; s < 4; ++s)
                #pragma unroll
                for (int i = 0; i < 4; ++i)
                    bv[nxt][s][i] = *reinterpret_cast<const v2f*>(Brow[s] + k0 + 16 + i * 4);
        }
        #pragma unroll
        for (int i = 0; i < 4; ++i)
            #pragma unroll
            for (int s = 0; s < 4; ++s)
                acc[s] = __builtin_amdgcn_wmma_f32_16x16x4_f32(
                    false, a[cur][i], false, bv[cur][s][i], (short)0, acc[s], false, false);
    }

    #pragma unroll
    for (int v = 0; v < 8; ++v) {
        int m = m0 + v + 8 * half;
        #pragma unroll
        for (int s = 0; s < 4; ++s) {
            int c = c0 + s * 16 + l15;
            KVb[((size_t)b * Dd + m) * Cch + c] = acc[s][v];
        }
    }
}

// ---------------------------------------------------------------------------
// Kernel 3: Ksum[b][m] = sum_n K[b][m][n] + EPS
// ---------------------------------------------------------------------------
__global__ void __launch_bounds__(256) ksum_kernel(
    const float* __restrict__ Kb, float* __restrict__ Ksum)
{
    __shared__ float red[256];
    const int bm = blockIdx.x;
    float s = 0.f;
    const float* row = Kb + (size_t)bm * Nsp;
    for (int i = threadIdx.x; i < Nsp; i += 256) s += row[i];
    red[threadIdx.x] = s;
    __syncthreads();
    for (int st = 128; st > 0; st >>= 1) {
        if (threadIdx.x < st) red[threadIdx.x] += red[threadIdx.x + st];
        __syncthreads();
    }
    if (threadIdx.x == 0) Ksum[bm] = red[0] + 1e-6f;
}

// ---------------------------------------------------------------------------
// Kernel 4: norm[b][n] = 1 / dot(Q[b][n][:], Ksum[b][:])
// ---------------------------------------------------------------------------
__global__ void __launch_bounds__(256) norm_kernel(
    const float* __restrict__ Qb, const float* __restrict__ Ksum,
    float* __restrict__ normb)
{
    const int i = blockIdx.x * 256 + threadIdx.x;
    const int b = i >> 12;
    const float* q  = Qb + (size_t)i * Dd;
    const float* ks = Ksum + b * Dd;
    float s = 0.f;
    #pragma unroll 8
    for (int m = 0; m < Dd; ++m) s += q[m] * ks[m];
    normb[i] = 1.0f / s;
}

// ---------------------------------------------------------------------------
// Kernel 5: out[b][c][n] = x + gamma * norm[n] * (Q[n,:] . KV[:,c]).
// Wave = 16(n) x 64(c). KV tile (64x64) in pair-interleaved LDS; Q fragments
// fully in registers (K=64). B fragments register double-buffered.
// Lane owns 8 consecutive n at fixed c -> b128 residual loads + stores.
// ---------------------------------------------------------------------------
__global__ void __launch_bounds__(128) out_kernel(
    const float* __restrict__ X, const float* __restrict__ Qb,
    const float* __restrict__ KVb, const float* __restrict__ normb,
    const float* __restrict__ gamma, float* __restrict__ out)
{
    __shared__ float KVs[32 * PROW];            // 20 KB

    const int tid  = threadIdx.x;
    const int wave = tid >> 5;
    const int lane = tid & 31;
    const int half = lane >> 4;
    const int l15  = lane & 15;

    int bid = blockIdx.x;
    const int cg  = bid % 8;  bid /= 8;
    const int tN4 = bid % 64;
    const int b   = bid / 64;

    const int n0 = (tN4 * 4 + wave) * 16;
    const int c0 = cg * 64;

    #pragma unroll
    for (int j = 0; j < 4; ++j) {
        int blk = tid + j * 128;                // 0..511
        int p   = blk >> 4;
        int cq  = (blk & 15) * 4;
        float4 r0 = *reinterpret_cast<const float4*>(
            KVb + ((size_t)b * Dd + 2*p    ) * Cch + c0 + cq);
        float4 r1 = *reinterpret_cast<const float4*>(
            KVb + ((size_t)b * Dd + 2*p + 1) * Cch + c0 + cq);
        float* dst = &KVs[p * PROW + cq * 2];
        *reinterpret_cast<float4*>(dst)     = make_float4(r0.x, r1.x, r0.y, r1.y);
        *reinterpret_cast<float4*>(dst + 4) = make_float4(r0.z, r1.z, r0.w, r1.w);
    }
    __syncthreads();

    const float* Arow = Qb + ((size_t)b * Nsp + n0 + l15) * Dd;
    v2f areg[16];
    #pragma unroll
    for (int kk = 0; kk < 16; ++kk)
        areg[kk] = *reinterpret_cast<const v2f*>(Arow + kk * 4 + 2 * half);

    v8f acc[4];
    #pragma unroll
    for (int s = 0; s < 4; ++s) acc[s] = (v8f){0.f,0.f,0.f,0.f,0.f,0.f,0.f,0.f};

    const int coff = l15 * 2;
    v2f bfrag[4];
    #pragma unroll
    for (int s = 0; s < 4; ++s)
        bfrag[s] = *reinterpret_cast<const v2f*>(&KVs[half * PROW + s * 32 + coff]);

    #pragma unroll
    for (int kk = 0; kk < 16; ++kk) {
        v2f bnext[4];
        if (kk < 15) {
            const int pn = (kk + 1) * 2 + half;
            #pragma unroll
            for (int s = 0; s < 4; ++s)
                bnext[s] = *reinterpret_cast<const v2f*>(&KVs[pn * PROW + s * 32 + coff]);
        }
        #pragma unroll
        for (int s = 0; s < 4; ++s)
            acc[s] = __builtin_amdgcn_wmma_f32_16x16x4_f32(
                false, areg[kk], false, bfrag[s], (short)0, acc[s], false, false);
        if (kk < 15) {
            #pragma unroll
            for (int s = 0; s < 4; ++s) bfrag[s] = bnext[s];
        }
    }

    const float g = gamma[0];
    const int nbase = n0 + 8 * half;
    float nv[8];
    #pragma unroll
    for (int v = 0; v < 8; ++v) nv[v] = normb[b * Nsp + nbase + v];

    #pragma unroll
    for (int s = 0; s < 4; ++s) {
        int c = c0 + s * 16 + l15;
        size_t base = ((size_t)b * Cch + c) * Nsp + nbase;
        float4 x0 = *reinterpret_cast<const float4*>(X + base);
        float4 x1 = *reinterpret_cast<const float4*>(X + base + 4);
        float4 o0, o1;
        o0.x = x0.x + g * acc[s][0] * nv[0];
        o0.y = x0.y + g * acc[s][1] * nv[1];
        o0.z = x0.z + g * acc[s][2] * nv[2];
        o0.w = x0.w + g * acc[s][3] * nv[3];
        o1.x = x1.x + g * acc[s][4] * nv[4];
        o1.y = x1.y + g * acc[s][5] * nv[5];
        o1.z = x1.z + g * acc[s][6] * nv[6];
        o1.w = x1.w + g * acc[s][7] * nv[7];
        *reinterpret_cast<float4*>(out + base)     = o0;
        *reinterpret_cast<float4*>(out + base + 4) = o1;
    }
}

// ---------------------------------------------------------------------------
extern "C" void kernel_launch(void* const* d_in, const int* in_sizes, int n_in,
                              void* d_out, int out_size, void* d_ws, size_t ws_size,
                              hipStream_t stream)
{
    const float* x     = (const float*)d_in[0];
    const float* Wq    = (const float*)d_in[1];
    const float* bq    = (const float*)d_in[2];
    const float* Wk    = (const float*)d_in[3];
    const float* bk    = (const float*)d_in[4];
    const float* Wv    = (const float*)d_in[5];
    const float* bv    = (const float*)d_in[6];
    const float* gamma = (const float*)d_in[7];
    float* out = (float*)d_out;

    float* ws    = (float*)d_ws;
    float* Qb    = ws;                                   // [B,N,d]   8 MB
    float* Kb    = Qb    + (size_t)Bsz * Nsp * Dd;       // [B,d,N]   8 MB
    float* Vb    = Kb    + (size_t)Bsz * Dd * Nsp;       // [B,C,N]  64 MB
    float* KVb   = Vb    + (size_t)Bsz * Cch * Nsp;      // [B,d,C]   1 MB
    float* Ksumb = KVb   + (size_t)Bsz * Dd * Cch;       // [B,d]
    float* normb = Ksumb + (size_t)Bsz * Dd;             // [B,N]

    qkv_kernel<<<Bsz * 64 * 10, 128, 0, stream>>>(
        x, Wq, bq, Wk, bk, Wv, bv, Qb, Kb, Vb);

    kv_kernel<<<(Bsz * 4 * 8) / 4, 128, 0, stream>>>(Kb, Vb, KVb);

    ksum_kernel<<<Bsz * Dd, 256, 0, stream>>>(Kb, Ksumb);

    norm_kernel<<<(Bsz * Nsp) / 256, 256, 0, stream>>>(Qb, Ksumb, normb);

    out_kernel<<<Bsz * 64 * 8, 128, 0, stream>>>(
        x, Qb, KVb, normb, gamma, out);
}